// ST_Mamba_VAR_2405181686044
// MI455X (gfx1250) — compile-verified
//
#include <hip/hip_runtime.h>
#include <hip/hip_bf16.h>
#include <math.h>

typedef __attribute__((ext_vector_type(2))) float v2f;
typedef __attribute__((ext_vector_type(4))) float v4f;
typedef __attribute__((ext_vector_type(8))) float v8f;

#define T_TOK 1024   // B*K
#define DM    512    // D_MODEL
#define DI    1024   // D_INNER
#define P_DIM 256
#define KSEQ  64
#define NB    16     // batch

__device__ __forceinline__ float act_apply(float v, int act) {
  if (act == 1) return 0.5f * v * (1.0f + erff(v * 0.70710678118654752f)); // exact gelu
  if (act == 2) return 1.0f / (1.0f + expf(-v));                          // sigmoid
  if (act == 3) return (v > 20.0f) ? v : log1pf(expf(v));                 // softplus
  return v;
}

// ---------------------------------------------------------------------------
// Wave-per-tile fp32 WMMA GEMM:  C[M,N] (+)= act( A[M,K] @ W[N,K]^T + bias )
// One wave (32 lanes) computes a 16x16 tile via V_WMMA_F32_16X16X4_F32.
// A frag (16x4):  lanes 0-15 rows M, VGPR0=K0 VGPR1=K1; lanes 16-31 K2/K3.
// B frag (4x16):  lanes 0-15 cols N, VGPR0=K0 VGPR1=K1; lanes 16-31 K2/K3.
// C/D (16x16):    VGPR v -> row tm + v + 8*(lane>=16), col tn + lane%16.
// ---------------------------------------------------------------------------
__global__ __launch_bounds__(32) void wmma_gemm_f32(
    const float* __restrict__ A, int lda,
    const float* __restrict__ W, int ldw,
    const float* __restrict__ bias,
    float* __restrict__ C, int ldc,
    int ntiles_n, int Kdim, int act, int accum)
{
  const int tile = blockIdx.x;
  const int tm = (tile / ntiles_n) << 4;
  const int tn = (tile % ntiles_n) << 4;
  const int lane = threadIdx.x;
  const int half = lane >> 4;   // 0: K pair {0,1}; 1: K pair {2,3}
  const int lid  = lane & 15;

  const float* arow = A + (size_t)(tm + lid) * lda + 2 * half;
  const float* wrow = W + (size_t)(tn + lid) * ldw + 2 * half;

  v8f acc = {0.f, 0.f, 0.f, 0.f, 0.f, 0.f, 0.f, 0.f};
  for (int k = 0; k < Kdim; k += 4) {
    v2f a; a.x = arow[k]; a.y = arow[k + 1];
    v2f b; b.x = wrow[k]; b.y = wrow[k + 1];
    acc = __builtin_amdgcn_wmma_f32_16x16x4_f32(
        /*neg_a=*/false, a, /*neg_b=*/false, b,
        /*c_mod=*/(short)0, acc, /*reuse_a=*/false, /*reuse_b=*/false);
  }

  const int c = tn + lid;
  const float bv = bias ? bias[c] : 0.f;
#pragma unroll
  for (int v = 0; v < 8; ++v) {
    const int r = tm + v + 8 * half;
    float val = acc[v] + bv;
    val = act_apply(val, act);
    float* p = C + (size_t)r * ldc + c;
    if (accum) val += *p;
    *p = val;
  }
}

// rmsnorm over rows of DM=512: y = x * rsqrt(mean(x^2)+eps) * w
__global__ __launch_bounds__(256) void rmsnorm_kernel(
    const float* __restrict__ x, const float* __restrict__ w, float* __restrict__ y)
{
  __shared__ float red[256];
  const int t = blockIdx.x, i = threadIdx.x;
  const float* xr = x + (size_t)t * DM;
  const float a = xr[i], b = xr[i + 256];
  red[i] = a * a + b * b;
  __syncthreads();
  for (int s = 128; s > 0; s >>= 1) { if (i < s) red[i] += red[i + s]; __syncthreads(); }
  const float scale = rsqrtf(red[0] / (float)DM + 1e-5f);
  float* yr = y + (size_t)t * DM;
  yr[i]       = a * scale * w[i];
  yr[i + 256] = b * scale * w[i + 256];
}

// h = rmsnorm(H2 * G, w)   (sp-front gate + norm)
__global__ __launch_bounds__(256) void gatenorm_kernel(
    const float* __restrict__ h2, const float* __restrict__ g,
    const float* __restrict__ w, float* __restrict__ y)
{
  __shared__ float red[256];
  const int t = blockIdx.x, i = threadIdx.x;
  const size_t base = (size_t)t * DM;
  const float a = h2[base + i]       * g[base + i];
  const float b = h2[base + i + 256] * g[base + i + 256];
  red[i] = a * a + b * b;
  __syncthreads();
  for (int s = 128; s > 0; s >>= 1) { if (i < s) red[i] += red[i + s]; __syncthreads(); }
  const float scale = rsqrtf(red[0] / (float)DM + 1e-5f);
  y[base + i]       = a * scale * w[i];
  y[base + i + 256] = b * scale * w[i + 256];
}

// Depthwise causal conv (width 4) + silu. xr is T x 2048 (x in cols 0..1023).
__global__ __launch_bounds__(256) void conv_silu_kernel(
    const float* __restrict__ xr, const float* __restrict__ cw,
    const float* __restrict__ cb, float* __restrict__ xc)
{
  const int idx = blockIdx.x * 256 + threadIdx.x; // B*DI threads
  const int d = idx & (DI - 1);
  const int b = idx >> 10;
  const v4f wv = *(const v4f*)(cw + (size_t)d * 4);
  const float bias = cb[d];
  float h0 = 0.f, h1 = 0.f, h2 = 0.f;
  for (int k = 0; k < KSEQ; ++k) {
    const size_t t = (size_t)b * KSEQ + k;
    const float xk = xr[t * 2048 + d];
    float v = wv.x * h0 + wv.y * h1 + wv.z * h2 + wv.w * xk + bias;
    v = v / (1.f + expf(-v)); // silu
    xc[t * DI + d] = v;
    h0 = h1; h1 = h2; h2 = xk;
  }
}

// Selective scan per (b,d) channel, N=16 state, + D skip + silu(res) gate.
__global__ __launch_bounds__(256) void scan_kernel(
    const float* __restrict__ xc, const float* __restrict__ delta,
    const float* __restrict__ xdbl, const float* __restrict__ A_log,
    const float* __restrict__ Dp, const float* __restrict__ xr,
    float* __restrict__ yg)
{
  const int idx = blockIdx.x * 256 + threadIdx.x;
  const int d = idx & (DI - 1);
  const int b = idx >> 10;
  float Ad[16];
#pragma unroll
  for (int n = 0; n < 16; ++n) Ad[n] = -expf(A_log[d * 16 + n]);
  const float Dd = Dp[d];
  float st[16];
#pragma unroll
  for (int n = 0; n < 16; ++n) st[n] = 0.f;
  for (int k = 0; k < KSEQ; ++k) {
    const size_t t = (size_t)b * KSEQ + k;
    const float dt = delta[t * DI + d];
    const float u  = xc[t * DI + d];
    const float* bm = xdbl + t * 64 + 32;
    const float* cm = xdbl + t * 64 + 48;
    float y = 0.f;
#pragma unroll
    for (int n = 0; n < 16; ++n) {
      const float da = expf(dt * Ad[n]);
      st[n] = da * st[n] + dt * bm[n] * u;
      y += st[n] * cm[n];
    }
    y += u * Dd;
    const float r = xr[t * 2048 + DI + d];
    y *= r / (1.f + expf(-r)); // * silu(res)
    yg[t * DI + d] = y;
  }
}

// ---------------------------------------------------------------------------
// Per-token outer product coeffs[t,p,q] = sum_r U[p,r]V[q,r].
// 256 MB stream-out: each thread writes 4 consecutive q's with non-temporal
// b128 stores (512 B per wave-store, NT hint keeps the hot buffers in L2).
// Also produces s[t,r] = sum_q V[q,r] * x[t,q] for the preds contraction.
// ---------------------------------------------------------------------------
__global__ __launch_bounds__(256) void coeffs_kernel(
    const float* __restrict__ out, const float* __restrict__ x,
    float* __restrict__ coeffs, float* __restrict__ svec)
{
  __shared__ float U[256][4];
  __shared__ float V[256][4];
  __shared__ float red[256];
  const int t = blockIdx.x, q = threadIdx.x;
  const v4f* row4 = (const v4f*)(out + (size_t)t * 2048);
  const v4f uq = row4[q];
  const v4f vq = row4[256 + q];
  U[q][0] = uq.x; U[q][1] = uq.y; U[q][2] = uq.z; U[q][3] = uq.w;
  V[q][0] = vq.x; V[q][1] = vq.y; V[q][2] = vq.z; V[q][3] = vq.w;
  const float inq = x[(size_t)t * P_DIM + q];
  __syncthreads();

  // s[t,r] = sum_q V[q][r] * x[t,q]
#pragma unroll
  for (int r = 0; r < 4; ++r) {
    red[q] = V[q][r] * inq;
    __syncthreads();
    for (int s = 128; s > 0; s >>= 1) { if (q < s) red[q] += red[q + s]; __syncthreads(); }
    if (q == 0) svec[(size_t)t * 4 + r] = red[0];
    __syncthreads();
  }

  // Stream-out: thread handles row p0 + sub, columns qc..qc+3 (4 rows/iter).
  const int sub = q >> 6;        // 0..3
  const int qc  = (q & 63) * 4;  // 0,4,...,252
  float vr[4][4];
#pragma unroll
  for (int j = 0; j < 4; ++j)
#pragma unroll
    for (int r = 0; r < 4; ++r) vr[j][r] = V[qc + j][r];

  float* base = coeffs + (size_t)t * 65536;
  for (int p0 = 0; p0 < 256; p0 += 4) {
    const int p = p0 + sub;
    const float u0 = U[p][0], u1 = U[p][1], u2 = U[p][2], u3 = U[p][3];
    v4f o;
    o.x = u0 * vr[0][0] + u1 * vr[0][1] + u2 * vr[0][2] + u3 * vr[0][3];
    o.y = u0 * vr[1][0] + u1 * vr[1][1] + u2 * vr[1][2] + u3 * vr[1][3];
    o.z = u0 * vr[2][0] + u1 * vr[2][1] + u2 * vr[2][2] + u3 * vr[2][3];
    o.w = u0 * vr[3][0] + u1 * vr[3][1] + u2 * vr[3][2] + u3 * vr[3][3];
    __builtin_nontemporal_store(o, (v4f*)(base + (size_t)p * 256 + qc));
  }
}

// preds[b,p] = sum_k sum_r U[t,p,r] * s[t,r]
__global__ __launch_bounds__(256) void preds_kernel(
    const float* __restrict__ out, const float* __restrict__ svec, float* __restrict__ preds)
{
  const int b = blockIdx.x, p = threadIdx.x;
  float acc = 0.f;
  for (int k = 0; k < KSEQ; ++k) {
    const size_t t = (size_t)b * KSEQ + k;
    const v4f U = *(const v4f*)(out + t * 2048 + (size_t)p * 4);
    const v4f s = *(const v4f*)(svec + t * 4);
    acc += U.x * s.x + U.y * s.y + U.z * s.z + U.w * s.w;
  }
  preds[(size_t)b * P_DIM + p] = acc;
}

extern "C" void kernel_launch(void* const* d_in, const int* in_sizes, int n_in,
                              void* d_out, int out_size, void* d_ws, size_t ws_size,
                              hipStream_t stream)
{
  (void)in_sizes; (void)n_in; (void)out_size; (void)ws_size;
  const float* inputs     = (const float*)d_in[0];
  const float* sp_lin1_w  = (const float*)d_in[1];
  const float* sp_lin1_b  = (const float*)d_in[2];
  const float* sp_lin2_w  = (const float*)d_in[3];
  const float* sp_lin2_b  = (const float*)d_in[4];
  const float* sp_gate_w  = (const float*)d_in[5];
  const float* sp_gate_b  = (const float*)d_in[6];
  const float* sp_norm_w  = (const float*)d_in[7];
  const float* in_proj_w  = (const float*)d_in[8];
  const float* conv_w     = (const float*)d_in[9];
  const float* conv_b     = (const float*)d_in[10];
  const float* x_proj_w   = (const float*)d_in[11];
  const float* dt_proj_w  = (const float*)d_in[12];
  const float* dt_proj_b  = (const float*)d_in[13];
  const float* A_log      = (const float*)d_in[14];
  const float* Dp         = (const float*)d_in[15];
  const float* out_proj_w = (const float*)d_in[16];
  const float* blk_norm_w = (const float*)d_in[17];
  const float* post_norm_w= (const float*)d_in[18];
  const float* proj_w     = (const float*)d_in[19];

  char* ws = (char*)d_ws;
  const size_t MB = (size_t)1 << 20;
  float* h    = (float*)(ws + 0 * MB);   // T x 512
  float* hn   = (float*)(ws + 2 * MB);   // T x 512
  float* big  = (float*)(ws + 4 * MB);   // T x 2048 (xr / H2 / out)
  float* xc   = (float*)(ws + 12 * MB);  // T x 1024 (also G in front)
  float* xdbl = (float*)(ws + 16 * MB);  // T x 64
  float* delta= (float*)(ws + 17 * MB);  // T x 1024
  float* yg   = (float*)(ws + 21 * MB);  // T x 1024
  float* svec = (float*)(ws + 25 * MB);  // T x 4

  float* preds  = (float*)d_out;
  float* coeffs = (float*)d_out + (size_t)NB * P_DIM;

  const int MT = T_TOK / 16; // 64 M-tiles

  // ---- front: H1 = gelu(X @ W1^T + b1) -> hn
  wmma_gemm_f32<<<MT * (DM / 16), 32, 0, stream>>>(
      inputs, P_DIM, sp_lin1_w, P_DIM, sp_lin1_b, hn, DM, DM / 16, P_DIM, /*act*/1, 0);
  // ---- G = sigmoid(X @ Wg^T + bg) -> xc
  wmma_gemm_f32<<<MT * (DM / 16), 32, 0, stream>>>(
      inputs, P_DIM, sp_gate_w, P_DIM, sp_gate_b, xc, DM, DM / 16, P_DIM, /*act*/2, 0);
  // ---- H2 = H1 @ W2^T + b2 -> big
  wmma_gemm_f32<<<MT * (DM / 16), 32, 0, stream>>>(
      hn, DM, sp_lin2_w, DM, sp_lin2_b, big, DM, DM / 16, DM, /*act*/0, 0);
  // ---- h = rmsnorm(H2 * G)
  gatenorm_kernel<<<T_TOK, 256, 0, stream>>>(big, xc, sp_norm_w, h);

  for (int l = 0; l < 4; ++l) {
    // hn = rmsnorm(h, blk_norm_w[l])
    rmsnorm_kernel<<<T_TOK, 256, 0, stream>>>(h, blk_norm_w + (size_t)l * DM, hn);
    // big(xr) = hn @ in_proj_w[l]^T   (T x 2048)
    wmma_gemm_f32<<<MT * (2048 / 16), 32, 0, stream>>>(
        hn, DM, in_proj_w + (size_t)l * 2048 * DM, DM, nullptr,
        big, 2048, 2048 / 16, DM, 0, 0);
    // xc = silu(causal depthwise conv(xr[:, :1024]))
    conv_silu_kernel<<<(NB * DI) / 256, 256, 0, stream>>>(
        big, conv_w + (size_t)l * DI * 4, conv_b + (size_t)l * DI, xc);
    // xdbl = xc @ x_proj_w[l]^T   (T x 64)
    wmma_gemm_f32<<<MT * (64 / 16), 32, 0, stream>>>(
        xc, DI, x_proj_w + (size_t)l * 64 * DI, DI, nullptr,
        xdbl, 64, 64 / 16, DI, 0, 0);
    // delta = softplus(xdbl[:, :32] @ dt_proj_w[l]^T + dt_b)   (T x 1024)
    wmma_gemm_f32<<<MT * (DI / 16), 32, 0, stream>>>(
        xdbl, 64, dt_proj_w + (size_t)l * DI * 32, 32, dt_proj_b + (size_t)l * DI,
        delta, DI, DI / 16, 32, /*act*/3, 0);
    // yg = selective_scan(...) * silu(res)
    scan_kernel<<<(NB * DI) / 256, 256, 0, stream>>>(
        xc, delta, xdbl, A_log + (size_t)l * DI * 16, Dp + (size_t)l * DI, big, yg);
    // h += yg @ out_proj_w[l]^T
    wmma_gemm_f32<<<MT * (DM / 16), 32, 0, stream>>>(
        yg, DI, out_proj_w + (size_t)l * DM * DI, DI, nullptr,
        h, DM, DM / 16, DI, 0, /*accum*/1);
  }

  // hn = rmsnorm(h, post_norm_w)
  rmsnorm_kernel<<<T_TOK, 256, 0, stream>>>(h, post_norm_w, hn);
  // big(out) = hn @ proj_w^T   (T x 2048)
  wmma_gemm_f32<<<MT * (2048 / 16), 32, 0, stream>>>(
      hn, DM, proj_w, DM, nullptr, big, 2048, 2048 / 16, DM, 0, 0);
  // coeffs stream-out (256 MB, HBM-bound, NT b128 stores) + s vector
  coeffs_kernel<<<T_TOK, 256, 0, stream>>>(big, inputs, coeffs, svec);
  // preds contraction (never re-reads coeffs)
  preds_kernel<<<NB, 256, 0, stream>>>(big, svec, preds);
}